// Group_45397804319538
// MI455X (gfx1250) — compile-verified
//
#include <hip/hip_runtime.h>
#include <hip/hip_bf16.h>
#include <math.h>

#define B_DIM   16
#define N_PTS   8192
#define NGROUP  512
#define KNN     32
#define GTILE   16
#define CHUNK   512
#define FPS_THREADS 1024
#define KNN_THREADS 256
#define BIGF 3.0e38f

typedef float v2f __attribute__((ext_vector_type(2)));
typedef float v8f __attribute__((ext_vector_type(8)));

// ---------------------------------------------------------------------------
// Kernel 1: farthest point sampling. One block per batch. Points are staged
// once into LDS (96 KB SoA + 32 KB running distances fits CDNA5's 320 KB WGP
// LDS), so each of the 511 dependent steps touches only LDS. Argmax uses a
// wave32 shuffle butterfly: 2 barriers/step instead of 10.
// ---------------------------------------------------------------------------
__global__ __launch_bounds__(FPS_THREADS)
void fps_kernel(const float* __restrict__ xyz, float* __restrict__ out_center)
{
    __shared__ float sx[N_PTS], sy[N_PTS], sz[N_PTS];   // 96 KB point SoA
    __shared__ float sdist[N_PTS];                      // 32 KB running min d^2
    __shared__ float wval[FPS_THREADS / 32];
    __shared__ int   widx[FPS_THREADS / 32];
    __shared__ float slast[3];

    const int b   = blockIdx.x;
    const int tid = threadIdx.x;
    const float* pts = xyz + (size_t)b * N_PTS * 3;
    float* cen = out_center + (size_t)b * NGROUP * 3;

    // stage batch into LDS
    for (int i = tid; i < N_PTS; i += FPS_THREADS) {
        sx[i] = pts[i * 3 + 0];
        sy[i] = pts[i * 3 + 1];
        sz[i] = pts[i * 3 + 2];
        sdist[i] = BIGF;
    }
    __syncthreads();

    if (tid == 0) {
        slast[0] = sx[0]; slast[1] = sy[0]; slast[2] = sz[0];
        cen[0] = sx[0]; cen[1] = sy[0]; cen[2] = sz[0];
    }
    __syncthreads();

    for (int it = 1; it < NGROUP; ++it) {
        const float lx = slast[0], ly = slast[1], lz = slast[2];
        float bv = -1.0f; int bi = N_PTS;
        for (int i = tid; i < N_PTS; i += FPS_THREADS) {
            float dx = sx[i] - lx;
            float dy = sy[i] - ly;
            float dz = sz[i] - lz;
            float d  = dx * dx + dy * dy + dz * dz;
            float mi = fminf(sdist[i], d);
            sdist[i] = mi;
            if (mi > bv) { bv = mi; bi = i; }   // ascending i -> first max kept
        }
        // wave32 butterfly argmax, tie -> lowest index (matches jnp.argmax)
        for (int mask = 16; mask > 0; mask >>= 1) {
            float ov = __shfl_xor(bv, mask, 32);
            int   oi = __shfl_xor(bi, mask, 32);
            if (ov > bv || (ov == bv && oi < bi)) { bv = ov; bi = oi; }
        }
        if ((tid & 31) == 0) { wval[tid >> 5] = bv; widx[tid >> 5] = bi; }
        __syncthreads();
        if (tid < 32) {                         // 32 waves -> one wave reduces
            bv = wval[tid]; bi = widx[tid];
            for (int mask = 16; mask > 0; mask >>= 1) {
                float ov = __shfl_xor(bv, mask, 32);
                int   oi = __shfl_xor(bi, mask, 32);
                if (ov > bv || (ov == bv && oi < bi)) { bv = ov; bi = oi; }
            }
            if (tid == 0) {
                const float x = sx[bi], y = sy[bi], z = sz[bi];
                slast[0] = x; slast[1] = y; slast[2] = z;
                cen[it * 3 + 0] = x; cen[it * 3 + 1] = y; cen[it * 3 + 2] = z;
            }
        }
        __syncthreads();
    }
}

// ---------------------------------------------------------------------------
// Kernel 2: KNN + gather. One block per (batch, 16-center tile).
// Distance proxy D[m][j] = ||p_j||^2 - 2*c_m.p_j via V_WMMA_F32_16X16X4_F32
// (A row m = (-2cx,-2cy,-2cz,1), B col j = (px,py,pz,||p||^2)); ||c||^2 is
// row-constant so per-center top-k ordering is exact. Selection runs one
// 16-lane half-wave per center with shuffle butterflies and zero barriers
// (same-wave LDS ordering guarantees visibility), with ascending-extraction
// early exit against the running top-32.
// ---------------------------------------------------------------------------
__global__ __launch_bounds__(KNN_THREADS)
void knn_group_kernel(const float* __restrict__ xyz,
                      const float* __restrict__ centers,
                      float* __restrict__ out_nb)
{
    __shared__ float spx[CHUNK], spy[CHUNK], spz[CHUNK], spp[CHUNK]; // 8 KB
    __shared__ float sdist[GTILE][CHUNK];                            // 32 KB
    __shared__ float scx[GTILE], scy[GTILE], scz[GTILE];
    __shared__ float run_val[GTILE][KNN];                            // sorted asc
    __shared__ int   run_idx[GTILE][KNN];

    const int b    = blockIdx.y;
    const int g0   = blockIdx.x * GTILE;
    const int tid  = threadIdx.x;
    const int lane = tid & 31;
    const int wave = tid >> 5;          // 0..7

    const float* pts = xyz + (size_t)b * N_PTS * 3;
    const float* cen = centers + ((size_t)b * NGROUP + g0) * 3;

    if (tid < GTILE) {
        scx[tid] = cen[tid * 3 + 0];
        scy[tid] = cen[tid * 3 + 1];
        scz[tid] = cen[tid * 3 + 2];
    }
    for (int i = tid; i < GTILE * KNN; i += KNN_THREADS) {
        run_val[i >> 5][i & 31] = BIGF;
        run_idx[i >> 5][i & 31] = 0;
    }
    __syncthreads();

    // A matrix (16x4), WMMA A layout: lanes 0-15 hold K=0,1; lanes 16-31 K=2,3
    const int am = lane & 15;
    v2f A;
    if (lane < 16) { A.x = -2.0f * scx[am]; A.y = -2.0f * scy[am]; }
    else           { A.x = -2.0f * scz[am]; A.y = 1.0f; }

    for (int base = 0; base < N_PTS; base += CHUNK) {
        // stage chunk of points + squared norms into LDS
        for (int i = tid; i < CHUNK; i += KNN_THREADS) {
            float x = pts[(base + i) * 3 + 0];
            float y = pts[(base + i) * 3 + 1];
            float z = pts[(base + i) * 3 + 2];
            spx[i] = x; spy[i] = y; spz[i] = z;
            spp[i] = x * x + y * y + z * z;
        }
        __syncthreads();

        // prefetch next chunk into cache while we compute (global_prefetch_b8)
        if (base + CHUNK < N_PTS)
            __builtin_prefetch(pts + (size_t)(base + CHUNK) * 3 + tid * 6, 0, 3);

        // WMMA over CHUNK/16 column tiles, round-robined across the 8 waves
        for (int t = wave; t < CHUNK / 16; t += KNN_THREADS / 32) {
            const int col = t * 16 + am;
            // B matrix (4x16): lanes 0-15 rows K=0,1; lanes 16-31 rows K=2,3
            v2f Bv;
            if (lane < 16) { Bv.x = spx[col]; Bv.y = spy[col]; }
            else           { Bv.x = spz[col]; Bv.y = spp[col]; }
            v8f Dv = {};
            Dv = __builtin_amdgcn_wmma_f32_16x16x4_f32(
                     false, A, false, Bv, (short)0, Dv, false, false);
            // C/D layout: lanes 0-15 -> rows r, lanes 16-31 -> rows r+8
            const int rowBase = (lane < 16) ? 0 : 8;
#pragma unroll
            for (int r = 0; r < 8; ++r)
                sdist[rowBase + r][col] = Dv[r];
        }
        __syncthreads();

        // barrier-free per-half-wave selection: center m handled by lanes of
        // one half-wave; extract ascending, merge into running sorted top-32
        const int m   = tid >> 4;
        const int sub = tid & 15;
        for (int round = 0; round < KNN; ++round) {
            float bv = BIGF; int bi = 0x7fffffff;
            for (int j = sub; j < CHUNK; j += 16) {
                float v = sdist[m][j];
                if (v < bv) { bv = v; bi = j; }
            }
            // butterfly argmin across the 16 lanes (masks 1..8 stay in-half)
            for (int mask = 8; mask > 0; mask >>= 1) {
                float ov = __shfl_xor(bv, mask, 32);
                int   oi = __shfl_xor(bi, mask, 32);
                if (ov < bv || (ov == bv && oi < bi)) { bv = ov; bi = oi; }
            }
            // ascending extraction: once chunk-min can't beat list max, stop
            if (bv >= run_val[m][KNN - 1]) break;
            if (sub == 0) {
                sdist[m][bi] = BIGF;            // remove from pool
                const int gidx = base + bi;
                int pos = KNN - 1;
                while (pos > 0 && run_val[m][pos - 1] > bv) {
                    run_val[m][pos] = run_val[m][pos - 1];
                    run_idx[m][pos] = run_idx[m][pos - 1];
                    --pos;
                }
                run_val[m][pos] = bv;
                run_idx[m][pos] = gidx;
            }
        }
        __syncthreads();   // protect spx/sdist/run lists across chunks
    }

    // gather neighborhoods and subtract centers
    for (int p = tid; p < GTILE * KNN; p += KNN_THREADS) {
        const int m = p >> 5, k = p & 31;
        const int j = run_idx[m][k];
        const float nx = pts[j * 3 + 0] - scx[m];
        const float ny = pts[j * 3 + 1] - scy[m];
        const float nz = pts[j * 3 + 2] - scz[m];
        const size_t o = (((size_t)b * NGROUP + (g0 + m)) * KNN + k) * 3;
        out_nb[o + 0] = nx; out_nb[o + 1] = ny; out_nb[o + 2] = nz;
    }
}

// ---------------------------------------------------------------------------
extern "C" void kernel_launch(void* const* d_in, const int* in_sizes, int n_in,
                              void* d_out, int out_size, void* d_ws, size_t ws_size,
                              hipStream_t stream)
{
    (void)in_sizes; (void)n_in; (void)out_size; (void)d_ws; (void)ws_size;

    const float* xyz  = (const float*)d_in[0];
    float* out        = (float*)d_out;
    float* out_nb     = out;                                        // [B,G,K,3]
    float* out_center = out + (size_t)B_DIM * NGROUP * KNN * 3;     // [B,G,3]

    fps_kernel<<<B_DIM, FPS_THREADS, 0, stream>>>(xyz, out_center);

    dim3 grid(NGROUP / GTILE, B_DIM);
    knn_group_kernel<<<grid, KNN_THREADS, 0, stream>>>(xyz, out_center, out_nb);
}